// UnionKeypointCoverageLoss_90666759618723
// MI455X (gfx1250) — compile-verified
//
#include <hip/hip_runtime.h>
#include <stdint.h>

// Problem constants (from the reference): (32,1,512,512) f32, topk=180, tol=3.
#define B_    32
#define N_    (512 * 512)
#define K_    180
#define TOL_  3
#define CAP_  512           // >= 2*K (max union size)
#define TPB_  256           // 8 wave32 waves
#define HBINS_ 65536        // 256 KB LDS histogram -- only possible on gfx1250 (320KB WGP LDS)
#define TILES_ (N_ / (TPB_ * 4))   // 256 tiles of 4KB (float4 per thread)

// ---- gfx1250 async global->LDS path (ASYNCcnt), guarded so compile is safe ----
#if defined(__has_builtin)
#  if __has_builtin(__builtin_amdgcn_global_load_async_to_lds_b128) && \
      __has_builtin(__builtin_amdgcn_s_wait_asynccnt)
#    define ATH_ASYNC 1
#  endif
#endif
#ifndef ATH_ASYNC
#  define ATH_ASYNC 0
#endif

#if ATH_ASYNC
// Builtin signature (from hipcc diagnostic): first param is
// 'int __vector(4) __device__ *' == v4i in addrspace(1); second is the LDS
// (addrspace(3)) v4i pointer; then imm offset, imm cpol.
typedef int ai4 __attribute__((ext_vector_type(4)));
typedef __attribute__((address_space(1))) ai4* ai4_g;
typedef __attribute__((address_space(3))) ai4* ai4_l;

__device__ __forceinline__ void async_ld16(const void* gp, void* lp) {
  __builtin_amdgcn_global_load_async_to_lds_b128(
      (ai4_g)(uintptr_t)gp,
      (ai4_l)(uint32_t)(uintptr_t)lp,
      /*imm offset*/ 0, /*cpol*/ 0);
}
template <int CNT>
__device__ __forceinline__ void wait_async() {
  __builtin_amdgcn_s_wait_asynccnt(CNT);
}
#endif

// Order-preserving float -> uint key (larger float => larger key).
__device__ __forceinline__ unsigned fkey(float f) {
  unsigned u = __float_as_uint(f);
  return (u & 0x80000000u) ? ~u : (u | 0x80000000u);
}

__global__ void kp_init(float* out) { out[0] = 0.0f; }

// One block per (array, batch-row): exact k-th-largest key via 2-pass 16-bit radix
// select using a full 64K-bin LDS histogram + ds atomics. Streaming is double-
// buffered through LDS with async loads (ASYNCcnt) when available.
__global__ __launch_bounds__(TPB_)
void kp_select_thresholds(const float* __restrict__ rv, const float* __restrict__ ri,
                          const float* __restrict__ rf, unsigned* __restrict__ thrKey) {
  __shared__ unsigned hist[HBINS_];    // 256 KB
  __shared__ unsigned csum[TPB_];      // 1 KB
  __shared__ float4   stage[2][TPB_];  // 8 KB double buffer
  __shared__ unsigned sSel, sAbove;

  const int tid = threadIdx.x;
  const int arr = blockIdx.x >> 5;   // 0:rv 1:ri 2:rf
  const int b   = blockIdx.x & 31;
  const float* src = (arr == 0) ? rv : (arr == 1) ? ri : rf;
  const float4* gsrc = (const float4*)(src + (size_t)b * N_);
  unsigned kk = (arr == 2) ? (2u * K_) : (unsigned)K_;

  unsigned prefix16 = 0;

  for (int pass = 0; pass < 2; ++pass) {
    for (int i = tid; i < HBINS_; i += TPB_) hist[i] = 0u;
    __syncthreads();

    auto proc = [&](float4 v) {
      unsigned u[4] = {fkey(v.x), fkey(v.y), fkey(v.z), fkey(v.w)};
#pragma unroll
      for (int e = 0; e < 4; ++e) {
        if (pass == 0) {
          atomicAdd(&hist[u[e] >> 16], 1u);                 // ds_add_u32
        } else if ((u[e] >> 16) == prefix16) {
          atomicAdd(&hist[u[e] & 0xFFFFu], 1u);
        }
      }
    };

#if ATH_ASYNC
    // Double-buffered async global->LDS streaming; each lane owns its slot, so
    // per-wave s_wait_asynccnt is the only ordering needed (no barrier).
    async_ld16(&gsrc[tid], &stage[0][tid]);
    for (int t = 0; t < TILES_ - 1; ++t) {
      async_ld16(&gsrc[(t + 1) * TPB_ + tid], &stage[(t + 1) & 1][tid]);
      wait_async<1>();                 // tile t landed in LDS
      proc(stage[t & 1][tid]);
    }
    wait_async<0>();
    proc(stage[(TILES_ - 1) & 1][tid]);
#else
    for (int t = 0; t < TILES_; ++t) proc(gsrc[t * TPB_ + tid]);
#endif
    __syncthreads();

    // Per-thread partial sums of 256-bin chunks, then serial suffix scan by t0
    // (256+256 iterations: negligible).
    unsigned s = 0;
    for (int j = 0; j < 256; ++j) s += hist[tid * 256 + j];
    csum[tid] = s;
    __syncthreads();

    if (tid == 0) {
      unsigned cum = 0;
      int cSel = 0;
      for (int c = 255; c >= 0; --c) {
        unsigned h = csum[c];
        if (cum + h >= kk) { cSel = c; break; }
        cum += h;
      }
      int bSel = 0;
      for (int j = 255; j >= 0; --j) {
        unsigned h = hist[cSel * 256 + j];
        if (cum + h >= kk) { bSel = j; break; }
        cum += h;
      }
      sSel   = (unsigned)(cSel * 256 + bSel);  // selected bin
      sAbove = cum;                            // strictly-greater count
    }
    __syncthreads();
    if (pass == 0) { prefix16 = sSel; kk -= sAbove; }
    __syncthreads();
  }
  if (tid == 0) thrKey[arr * B_ + b] = (prefix16 << 16) | sSel;
}

// One block per batch row: compact keypoint indices into small lists.
__global__ __launch_bounds__(TPB_)
void kp_build_lists(const float* __restrict__ rv, const float* __restrict__ ri,
                    const float* __restrict__ rf, const unsigned* __restrict__ thrKey,
                    unsigned* __restrict__ srcCnt, unsigned* __restrict__ fuCnt,
                    int* __restrict__ srcList, int* __restrict__ fuList) {
  const int b = blockIdx.x;
  if (threadIdx.x == 0) { srcCnt[b] = 0u; fuCnt[b] = 0u; }  // re-zeroed every call
  __syncthreads();
  const unsigned tv = thrKey[0 * B_ + b];
  const unsigned ti = thrKey[1 * B_ + b];
  const unsigned tf = thrKey[2 * B_ + b];
  const float* pv = rv + (size_t)b * N_;
  const float* pi = ri + (size_t)b * N_;
  const float* pf = rf + (size_t)b * N_;
  for (int i = threadIdx.x; i < N_; i += TPB_) {
    if (i + 8 * TPB_ < N_) {                       // -> global_prefetch_b8
      __builtin_prefetch(pv + i + 8 * TPB_, 0, 0);
      __builtin_prefetch(pi + i + 8 * TPB_, 0, 0);
      __builtin_prefetch(pf + i + 8 * TPB_, 0, 0);
    }
    unsigned kv = fkey(pv[i]), ki = fkey(pi[i]), kf = fkey(pf[i]);
    if (kv >= tv || ki >= ti) {
      unsigned p = atomicAdd(&srcCnt[b], 1u);
      if (p < CAP_) srcList[b * CAP_ + p] = i;
    }
    if (kf >= tf) {
      unsigned p = atomicAdd(&fuCnt[b], 1u);
      if (p < CAP_) fuList[b * CAP_ + p] = i;
    }
  }
}

// One block per batch row: 7x7 dilation == Chebyshev distance <= TOL against the
// compact fused list. O(ns*nf) <= 360*360 per row.
__global__ __launch_bounds__(TPB_)
void kp_coverage(const unsigned* __restrict__ srcCnt, const unsigned* __restrict__ fuCnt,
                 const int* __restrict__ srcList, const int* __restrict__ fuList,
                 float* __restrict__ out) {
  __shared__ int fy[CAP_], fx[CAP_];
  __shared__ unsigned hitSum;
  const int b = blockIdx.x;
  unsigned ns = srcCnt[b]; ns = ns < CAP_ ? ns : CAP_;
  unsigned nf = fuCnt[b];  nf = nf < CAP_ ? nf : CAP_;
  if (threadIdx.x == 0) hitSum = 0u;
  for (unsigned j = threadIdx.x; j < nf; j += TPB_) {
    int id = fuList[b * CAP_ + j];
    fy[j] = id >> 9;
    fx[j] = id & 511;
  }
  __syncthreads();
  unsigned local = 0;
  for (unsigned s = threadIdx.x; s < ns; s += TPB_) {
    int id = srcList[b * CAP_ + s];
    int y = id >> 9, x = id & 511;
    int hit = 0;
    for (unsigned j = 0; j < nf; ++j) {
      int dy = y - fy[j]; dy = dy < 0 ? -dy : dy;
      int dx = x - fx[j]; dx = dx < 0 ? -dx : dx;
      if (dy <= TOL_ && dx <= TOL_) { hit = 1; break; }
    }
    local += (unsigned)hit;
  }
  atomicAdd(&hitSum, local);
  __syncthreads();
  if (threadIdx.x == 0) {
    float denom = (float)(ns > 0u ? ns : 1u);
    float cov = (float)hitSum / denom;
    atomicAdd(out, (1.0f - cov) * (1.0f / (float)B_));
  }
}

extern "C" void kernel_launch(void* const* d_in, const int* in_sizes, int n_in,
                              void* d_out, int out_size, void* d_ws, size_t ws_size,
                              hipStream_t stream) {
  (void)in_sizes; (void)n_in; (void)out_size; (void)ws_size;
  const float* rv = (const float*)d_in[0];
  const float* ri = (const float*)d_in[1];
  const float* rf = (const float*)d_in[2];
  float* out = (float*)d_out;

  // Workspace layout (~129 KB): thresholds | counts | index lists.
  unsigned* thr    = (unsigned*)d_ws;        // 3*32
  unsigned* srcCnt = thr + 3 * B_;           // 32
  unsigned* fuCnt  = srcCnt + B_;            // 32
  int* srcList = (int*)(fuCnt + B_);         // 32*512
  int* fuList  = srcList + B_ * CAP_;        // 32*512

  hipLaunchKernelGGL(kp_init, dim3(1), dim3(1), 0, stream, out);
  hipLaunchKernelGGL(kp_select_thresholds, dim3(3 * B_), dim3(TPB_), 0, stream,
                     rv, ri, rf, thr);
  hipLaunchKernelGGL(kp_build_lists, dim3(B_), dim3(TPB_), 0, stream,
                     rv, ri, rf, thr, srcCnt, fuCnt, srcList, fuList);
  hipLaunchKernelGGL(kp_coverage, dim3(B_), dim3(TPB_), 0, stream,
                     srcCnt, fuCnt, srcList, fuList, out);
}